// OSRiRoIAlign_489626272160
// MI455X (gfx1250) — compile-verified
//
#include <hip/hip_runtime.h>
#include <math.h>

// ---------------- problem constants (from reference) ----------------
#define OUT_SZ   7
#define G        2
#define P        (OUT_SZ * G)      // 14 sample points per axis
#define NPTS     (P * P)           // 196 sample points per roi
#define NORI     8
#define C_CH     256
#define H_F      128
#define W_F      128
#define PLANE    (H_F * W_F)
#define SCALE    0.25f
#define BINS     (OUT_SZ * OUT_SZ) // 49
#define TPB      256
#define NCHUNK   ((C_CH * BINS) / 4)   // 3136 float4 chunks per roi

#ifndef M_PI
#define M_PI 3.14159265358979323846
#endif

// gfx1250 async LDS->global store path (probe: guarded by __has_builtin)
#if defined(__AMDGCN__) && __has_builtin(__builtin_amdgcn_global_store_async_from_lds_b128)
#define USE_ASYNC_OUT 1
#else
#define USE_ASYNC_OUT 0
#endif

typedef float v4f __attribute__((ext_vector_type(4)));   // native vector (host+device OK)

#if defined(__AMDGCN__)
typedef int v4i __attribute__((ext_vector_type(4)));
typedef __attribute__((address_space(1))) v4i as1_v4i;   // global
typedef __attribute__((address_space(3))) v4i as3_v4i;   // LDS
#endif

__global__ __launch_bounds__(TPB)
void riroi_align_kernel(const float* __restrict__ feat,
                        const float* __restrict__ rois,
                        float* __restrict__ out)
{
    __shared__ int4   sOff[NPTS];            // 4 gather offsets per sample point
    __shared__ float4 sW[NPTS];              // 4 bilinear weights (x 1/4 mean, x valid)
    __shared__ float4 sOut4[NCHUNK];         // [C][49] staged output tile (50 KB)
    float* sOut = (float*)sOut4;

    const int roi_id = blockIdx.x;
    const int tid    = threadIdx.x;

    // ---- roi parameters (uniform -> scalar loads) ----
    const float* r = rois + (size_t)roi_id * 6;
    const int   b     = (int)r[0];
    const float cx    = r[1] * SCALE;
    const float cy    = r[2] * SCALE;
    const float rw    = fmaxf(r[3] * SCALE, 1.0f);
    const float rh    = fmaxf(r[4] * SCALE, 1.0f);
    const float theta = r[5];
    const float bw    = rw * (1.0f / OUT_SZ);
    const float bh    = rh * (1.0f / OUT_SZ);
    float st, ct;
    sincosf(theta, &st, &ct);

    // orientation interpolation parameters
    const float ind_f = theta * (float)(NORI / (2.0 * M_PI));
    const float ind0  = floorf(ind_f);
    const float l     = ind_f - ind0;
    const int   ind   = ((int)ind0) & (NORI - 1);

    // ---- phase 1: per-sample bilinear setup, shared across all channels ----
    if (tid < NPTS) {
        const int iy = tid / P;
        const int ix = tid - iy * P;
        const float off_y = (float)(iy >> 1) + 0.25f + 0.5f * (float)(iy & 1);
        const float off_x = (float)(ix >> 1) + 0.25f + 0.5f * (float)(ix & 1);
        const float yy = -0.5f * rh + off_y * bh;
        const float xx = -0.5f * rw + off_x * bw;
        const float x  = xx * ct + yy * st + cx;
        const float y  = yy * ct - xx * st + cy;

        const bool valid = (y > -1.0f) && (y < (float)H_F) &&
                           (x > -1.0f) && (x < (float)W_F);
        float xc = fmaxf(x, 0.0f);
        float yc = fmaxf(y, 0.0f);
        int x0 = (int)floorf(xc);
        int y0 = (int)floorf(yc);
        int x1, y1;
        if (x0 >= W_F - 1) { x0 = W_F - 1; x1 = W_F - 1; xc = (float)x0; }
        else               { x1 = x0 + 1; }
        if (y0 >= H_F - 1) { y0 = H_F - 1; y1 = H_F - 1; yc = (float)y0; }
        else               { y1 = y0 + 1; }
        const float lx = xc - (float)x0;
        const float ly = yc - (float)y0;
        const float wv = valid ? 0.25f : 0.0f;   // fold the g*g mean (exact pow2 scale)
        sOff[tid] = make_int4(y0 * W_F + x0, y0 * W_F + x1,
                              y1 * W_F + x0, y1 * W_F + x1);
        sW[tid] = make_float4(wv * (1.0f - ly) * (1.0f - lx),
                              wv * (1.0f - ly) * lx,
                              wv * ly * (1.0f - lx),
                              wv * ly * lx);
    }
    __syncthreads();

    // ---- phase 2: lane = channel; gather + pool + orientation mix ----
    const int c = tid;                                  // 256 threads == 256 channels
    const float* plane = feat + (size_t)b * (C_CH * PLANE) + (size_t)c * PLANE;

    const int o        = c & (NORI - 1);
    const int src      = (o - ind) & (NORI - 1);        // circular source orientation
    const int lanebase = tid & 24;                      // orientation-group base within wave32
    const int lsrc     = lanebase + src;
    const int lsrcp    = lanebase + ((src + 1) & (NORI - 1));
    const float oml    = 1.0f - l;

    #pragma unroll 2
    for (int bin = 0; bin < BINS; ++bin) {
        const int ph  = bin / OUT_SZ;
        const int pw  = bin - ph * OUT_SZ;
        const int sp0 = (ph * 2) * P + pw * 2;
        float acc = 0.0f;
        #pragma unroll
        for (int s = 0; s < 4; ++s) {
            const int sp = sp0 + (s >> 1) * P + (s & 1);
            const int4   ofs = sOff[sp];
            const float4 w   = sW[sp];
            acc = fmaf(w.x, plane[ofs.x], acc);
            acc = fmaf(w.y, plane[ofs.y], acc);
            acc = fmaf(w.z, plane[ofs.z], acc);
            acc = fmaf(w.w, plane[ofs.w], acc);
        }
        // orientation channel rotation via wave32 ds_bpermute (in-wave, no LDS trip)
        const float vs  = __shfl(acc, lsrc, 32);
        const float vsp = __shfl(acc, lsrcp, 32);
        sOut[c * BINS + bin] = oml * vs + l * vsp;
    }
    __syncthreads();

    // ---- phase 3: coalesced B128 writeout of the staged tile ----
    float* gout = out + (size_t)roi_id * (C_CH * BINS);
#if USE_ASYNC_OUT
    for (int k = tid; k < NCHUNK; k += TPB) {
        __builtin_amdgcn_global_store_async_from_lds_b128(
            (as1_v4i*)(gout + 4 * k), (as3_v4i*)(sOut4 + k),
            /*offset=*/0, /*cpol=*/0);
    }
#if __has_builtin(__builtin_amdgcn_s_wait_asynccnt)
    __builtin_amdgcn_s_wait_asynccnt(0);
#else
    asm volatile("s_wait_asynccnt 0" ::: "memory");
#endif
#else
    for (int k = tid; k < NCHUNK; k += TPB) {
        // streamed output: non-temporal so it doesn't evict the L2-resident features
        __builtin_nontemporal_store(*(const v4f*)&sOut4[k], (v4f*)(gout) + k);
    }
#endif
}

extern "C" void kernel_launch(void* const* d_in, const int* in_sizes, int n_in,
                              void* d_out, int out_size, void* d_ws, size_t ws_size,
                              hipStream_t stream) {
    const float* feat = (const float*)d_in[0];   // [N, 256, 128, 128] f32
    const float* rois = (const float*)d_in[1];   // [R, 6] f32
    float* out = (float*)d_out;                  // [R, 256, 7, 7] f32
    const int R = in_sizes[1] / 6;
    (void)n_in; (void)out_size; (void)d_ws; (void)ws_size;
    riroi_align_kernel<<<R, TPB, 0, stream>>>(feat, rois, out);
}